// DFAGNN_plus_max_68375879353076
// MI455X (gfx1250) — compile-verified
//
#include <hip/hip_runtime.h>
#include <hip/hip_bf16.h>

// Problem constants (match reference)
#define BB 2
#define NN 4096
#define EE 65536
#define MM 4
#define HH 256
#define MH (MM * HH)          // 1024 floats per (node) slab
#define ROWS (BB * NN * MM)   // 32768 rows for the final GEMM

typedef __attribute__((ext_vector_type(2))) float v2f;
typedef __attribute__((ext_vector_type(8))) float v8f;

// ---------------------------------------------------------------------------
// Kernel 0: initialize the segment-max accumulator X to -inf
// ---------------------------------------------------------------------------
__global__ void dfa_init_kernel(uint4* __restrict__ X4) {
    const unsigned NEG_INF = 0xFF800000u;
    uint4 v; v.x = NEG_INF; v.y = NEG_INF; v.z = NEG_INF; v.w = NEG_INF;
    size_t i = (size_t)blockIdx.x * blockDim.x + threadIdx.x;
    X4[i] = v;   // grid sized exactly: BB*NN*MM*HH/4 elements
}

// ---------------------------------------------------------------------------
// Kernel 1: per-edge scalar coefficient  coeff[b,e] = edge_fts[b,e,:].W + b
// One wave32 per edge: 8 fp32 FMAs per lane + shuffle reduction.
// ---------------------------------------------------------------------------
__global__ void dfa_coeff_kernel(const float* __restrict__ edge_fts,
                                 const float* __restrict__ edge_W,
                                 const float* __restrict__ edge_b,
                                 float* __restrict__ coeff) {
    const int wave = threadIdx.x >> 5;
    const int lane = threadIdx.x & 31;
    const size_t idx = (size_t)blockIdx.x * 8 + wave;      // flat edge in [0, BB*EE)
    const float* ef = edge_fts + idx * HH;
    float s = 0.f;
#pragma unroll
    for (int j = 0; j < 8; ++j) {
        int k = lane + 32 * j;
        s += ef[k] * edge_W[k];
    }
#pragma unroll
    for (int off = 16; off > 0; off >>= 1)
        s += __shfl_down(s, off, 32);
    if (lane == 0) coeff[idx] = s + edge_b[0];
}

// ---------------------------------------------------------------------------
// Kernel 2: scatter-max.  One block per edge; each edge reads a contiguous
// 4KB slab hint_state[b, src, :, :], scales by coeff, and max-reduces into
// X[b, tgt, :, :] with the native hardware float atomic
// (GLOBAL_ATOMIC_MAX_NUM_F32, non-returning).  X is pre-set to -inf and the
// inputs are NaN-free, so maximumNumber semantics match segment_max exactly.
// ---------------------------------------------------------------------------
__global__ void dfa_scatter_kernel(const int* __restrict__ cfg,     // [B,E,2]
                                   const float* __restrict__ hint,  // [B,N,M,H]
                                   const float* __restrict__ coeff, // [B*E]
                                   float* __restrict__ X) {         // [B,N,M,H]
    const int flat = blockIdx.x;        // [0, BB*EE)
    const int b = flat >> 16;           // EE = 65536
    const int s = cfg[2 * flat + 0];
    const int t = cfg[2 * flat + 1];
    const float c = coeff[flat];

    const float4* srcp = (const float4*)(hint + ((size_t)(b * NN + s)) * MH);
    float*        dstp = X + ((size_t)(b * NN + t)) * MH;

    float4 v = srcp[threadIdx.x];                 // coalesced 4KB slab read
    const int base = threadIdx.x * 4;
    unsafeAtomicMax(dstp + base + 0, c * v.x);
    unsafeAtomicMax(dstp + base + 1, c * v.y);
    unsafeAtomicMax(dstp + base + 2, c * v.z);
    unsafeAtomicMax(dstp + base + 3, c * v.w);
}

// ---------------------------------------------------------------------------
// Kernel 3: out = (node_fts + X) @ update_W + update_b  via fp32 WMMA.
// [32768 x 256] * [256 x 256].  128 threads = 4 waves per block.
// Block computes a 32-row x 64-col tile; each wave owns two stacked 16x16
// output tiles (rows r..r+15 and r+16..r+31) so every B fragment of update_W
// loaded from L2 feeds TWO V_WMMA_F32_16X16X4_F32 ops.
//
// Fragment layouts (ISA 7.12.2, 32-bit):
//   A 16x4 : lanes 0-15 -> M=lane, v0=K0 v1=K1 ; lanes 16-31 -> M=lane-16, v0=K2 v1=K3
//   B 4x16 : lanes 0-15 -> N=lane, v0=K0 v1=K1 ; lanes 16-31 -> N=lane-16, v0=K2 v1=K3
//   C/D    : VGPR j: lanes 0-15 -> (M=j, N=lane); lanes 16-31 -> (M=j+8, N=lane-16)
// ---------------------------------------------------------------------------
#define RT 32                                     // rows per workgroup
__global__ void __launch_bounds__(128)
dfa_gemm_kernel(const float* __restrict__ X,        // [ROWS, HH] aggregated max
                const float* __restrict__ node,     // [ROWS, HH]
                const float* __restrict__ W,        // [HH, HH] row-major (k, c)
                const float* __restrict__ bias,     // [HH]
                float* __restrict__ out) {          // [ROWS, HH]
    __shared__ float As[RT][HH + 4];                // pad: stride 260 -> banks 4 apart

    const int tid  = threadIdx.x;
    const int wave = tid >> 5;
    const int lane = tid & 31;
    const int rowBase = blockIdx.x * RT;

    // Cooperative load of the 32x256 A tile = node + X  (float4-coalesced)
    for (int i = tid; i < RT * (HH / 4); i += 128) {
        const int r  = i >> 6;          // 64 float4 per row
        const int c4 = (i & 63) * 4;
        const size_t g = (size_t)(rowBase + r) * HH + c4;
        float4 xv = *(const float4*)(X + g);
        float4 nv = *(const float4*)(node + g);
        As[r][c4 + 0] = xv.x + nv.x;
        As[r][c4 + 1] = xv.y + nv.y;
        As[r][c4 + 2] = xv.z + nv.z;
        As[r][c4 + 3] = xv.w + nv.w;
    }
    __syncthreads();

    const int cb   = blockIdx.y * 64 + wave * 16;   // this wave's column tile
    const int half = lane >> 4;                     // 0: K pair (0,1); 1: (2,3)
    const int l16  = lane & 15;

    v8f acc0 = {0.f, 0.f, 0.f, 0.f, 0.f, 0.f, 0.f, 0.f};
    v8f acc1 = {0.f, 0.f, 0.f, 0.f, 0.f, 0.f, 0.f, 0.f};

#pragma unroll 4
    for (int kk = 0; kk < HH; kk += 4) {
        const int k0 = kk + 2 * half;
        v2f bfrag;
        bfrag.x = W[(size_t)k0 * HH + cb + l16];
        bfrag.y = W[(size_t)(k0 + 1) * HH + cb + l16];
        v2f a0, a1;
        a0.x = As[l16][k0];
        a0.y = As[l16][k0 + 1];
        a1.x = As[16 + l16][k0];
        a1.y = As[16 + l16][k0 + 1];
        acc0 = __builtin_amdgcn_wmma_f32_16x16x4_f32(
            false, a0, false, bfrag, (short)0, acc0, false, false);
        acc1 = __builtin_amdgcn_wmma_f32_16x16x4_f32(
            false, a1, false, bfrag, (short)0, acc1, false, false);
    }

    const float bv = bias[cb + l16];
#pragma unroll
    for (int j = 0; j < 8; ++j) {
        const int r0 = rowBase + j + half * 8;           // tile 0: rows 0..15
        out[(size_t)r0 * HH + cb + l16] = acc0[j] + bv;
    }
#pragma unroll
    for (int j = 0; j < 8; ++j) {
        const int r1 = rowBase + 16 + j + half * 8;      // tile 1: rows 16..31
        out[(size_t)r1 * HH + cb + l16] = acc1[j] + bv;
    }
}

// ---------------------------------------------------------------------------
// Launcher
// ---------------------------------------------------------------------------
extern "C" void kernel_launch(void* const* d_in, const int* in_sizes, int n_in,
                              void* d_out, int out_size, void* d_ws, size_t ws_size,
                              hipStream_t stream) {
    const int*   cfg      = (const int*)  d_in[0];  // [B,E,2]
    const float* hint     = (const float*)d_in[1];  // [B,N,M,H]
    const float* node     = (const float*)d_in[2];  // [B,N,M,H]
    const float* edge_fts = (const float*)d_in[3];  // [B,E,H]
    const float* edge_W   = (const float*)d_in[4];  // [H,1]
    const float* edge_b   = (const float*)d_in[5];  // [1]
    const float* update_W = (const float*)d_in[6];  // [H,H]
    const float* update_b = (const float*)d_in[7];  // [H]
    float* out = (float*)d_out;                     // [B,N,M,H]

    // Workspace layout
    float* X     = (float*)d_ws;                          // 32 MB accumulator
    float* coeff = X + (size_t)BB * NN * MM * HH;         // 512 KB coefficients

    // 0) X = -inf
    {
        const int n4 = (BB * NN * MM * HH) / 4;           // uint4 elements
        dfa_init_kernel<<<n4 / 256, 256, 0, stream>>>((uint4*)X);
    }
    // 1) per-edge coefficients (8 waves / block, 1 wave per edge)
    dfa_coeff_kernel<<<(BB * EE) / 8, 256, 0, stream>>>(edge_fts, edge_W, edge_b, coeff);

    // 2) gather-scale-scatter-max (1 block per edge, native f32 max atomics)
    dfa_scatter_kernel<<<BB * EE, MH / 4, 0, stream>>>(cfg, hint, coeff, X);

    // 3) fused add + GEMM + bias with fp32 WMMA (2 row-tiles per wave)
    {
        dim3 grid(ROWS / RT, HH / 64);
        dfa_gemm_kernel<<<grid, 128, 0, stream>>>(X, node, update_W, update_b, out);
    }
}